// KalmanStaticFilter_39393440039122
// MI455X (gfx1250) — compile-verified
//
#include <hip/hip_runtime.h>
#include <hip/hip_bf16.h>

// KalmanStaticFilter on MI455X (gfx1250).
// Memory-bound affine scan: 3-pass chunked decomposition + TDM async tensor
// loads (tensor_load_to_lds / s_wait_tensorcnt) with double-buffered LDS
// staging; TDM pad feature gives stride-33 (bank-conflict-free) tile rows.

#define ROWS 4096
#define COLS 8192
#define CHUNKS 64
#define CHUNK_LEN 128            // COLS / CHUNKS
#define SUBT 32                  // columns per TDM subtile
#define NSUB (CHUNK_LEN / SUBT)  // 4
#define RPW 32                   // rows per wave
#define ROWBLOCKS (ROWS / RPW)   // 128
#define WPB 4                    // waves per block (128 threads)
#define LDS_STRIDE 33            // padded dword stride (TDM pad: 1 dword / 32)
#define BUF_FLOATS (RPW * LDS_STRIDE)   // 1056 floats per buffer

static __device__ __constant__ float kA = 0.85f;
static __device__ __constant__ float kB = 0.005f;

typedef unsigned int u32x4 __attribute__((ext_vector_type(4)));
typedef int          i32x4 __attribute__((ext_vector_type(4)));
typedef int          i32x8 __attribute__((ext_vector_type(8)));

// Issue one TDM load of a 32x32 f32 tile (row stride COLS in global) into LDS
// at lds_byte_addr, with per-row +1-dword padding done by the TDM itself.
__device__ __forceinline__ void tdm_load_tile(unsigned lds_byte_addr,
                                              const float* gtile) {
  unsigned long long ga = (unsigned long long)(const void*)gtile;

  // D# group 0: count=1 | lds_addr | global_addr(57b) | type=2
  u32x4 g0;
  g0[0] = 1u;                                   // count=1, is_restore=0, gather=0
  g0[1] = lds_byte_addr;                        // lds_addr (bytes)
  g0[2] = (unsigned)(ga & 0xFFFFFFFFull);       // global_addr[31:0]
  g0[3] = (unsigned)((ga >> 32) & 0x01FFFFFFull) | (2u << 30); // addr[56:32]|type=2

  // D# group 1
  i32x8 g1;
  g1[0] = (int)((2u << 16)        // data_size = 4B
              | (1u << 20)        // pad_enable
              | (4u << 22)        // pad_interval: 32 dwords
              | (0u << 25));      // pad_amount: 1 dword  -> LDS stride 33
  g1[1] = (int)(((unsigned)COLS & 0xFFFFu) << 16);            // tensor_dim0 lo16
  g1[2] = (int)(((unsigned)COLS >> 16) | (((unsigned)ROWS & 0xFFFFu) << 16));
  g1[3] = (int)(((unsigned)ROWS >> 16) | ((unsigned)SUBT << 16)); // tile_dim0=32
  g1[4] = (int)((unsigned)RPW);                               // tile_dim1=32, dim2=0
  g1[5] = (int)(unsigned)COLS;                                // tensor_dim0_stride
  g1[6] = 0;
  g1[7] = 0;

  i32x4 gz4 = {0, 0, 0, 0};                  // groups 2/3 unused (2D tile)
  i32x8 gz8 = {0, 0, 0, 0, 0, 0, 0, 0};      // extra operand (clang-23 form)
  __builtin_amdgcn_tensor_load_to_lds(g0, g1, gz4, gz4, gz8, 0);
}

__device__ __forceinline__ void wave_geometry(int& lane, int& r0, int& c0,
                                              int& chunk, unsigned& lds_addr,
                                              float* lds_base, float*& lds_wave) {
  int tid = (int)threadIdx.x;
  lane = tid & 31;
  int wid = __builtin_amdgcn_readfirstlane(tid >> 5);  // wave-uniform
  int gw = (int)blockIdx.x * WPB + wid;
  chunk = gw & (CHUNKS - 1);
  int rb = gw >> 6;  // / CHUNKS
  r0 = rb * RPW;
  c0 = chunk * CHUNK_LEN;
  float* mylds = lds_base + wid * (2 * BUF_FLOATS);
  lds_addr = (unsigned)(unsigned long long)(void*)mylds;  // low 32b = LDS offset
  lds_wave = mylds;
}

// Process one 32-row x 128-col chunk with a double-buffered TDM pipeline.
// EMIT=false: partial pass (s starts at 0, no output).  EMIT=true: final pass.
template <bool EMIT>
__device__ __forceinline__ void run_chunk(const float* gbase, float* lds_wave,
                                          unsigned lds_addr, int lane,
                                          float& p, float& v, float* optr) {
  tdm_load_tile(lds_addr, gbase);
  tdm_load_tile(lds_addr + BUF_FLOATS * 4u, gbase + SUBT);

#pragma unroll
  for (int k = 0; k < NSUB; ++k) {
    // In-order TDM completion: cnt<=1 means subtile k has landed.
    if (k == NSUB - 1) asm volatile("s_wait_tensorcnt 0x0" ::: "memory");
    else               asm volatile("s_wait_tensorcnt 0x1" ::: "memory");

    const float* rowp = lds_wave + (k & 1) * BUF_FLOATS + lane * LDS_STRIDE;
    float4 xb;
#pragma unroll
    for (int i = 0; i < SUBT; ++i) {
      float z = rowp[i];
      float diff = z - p;
      float x = fmaf(kA, diff, p);
      v = fmaf(kB, diff, v);
      p = x + v;
      if (EMIT) {
        if ((i & 3) == 0)      xb.x = x;
        else if ((i & 3) == 1) xb.y = x;
        else if ((i & 3) == 2) xb.z = x;
        else {
          xb.w = x;
          *(float4*)(optr + k * SUBT + (i - 3)) = xb;  // 16B-aligned
        }
      }
    }
    if (k + 2 < NSUB) {
      // All LDS reads of this buffer retired before the DMA overwrites it.
      asm volatile("s_wait_dscnt 0x0" ::: "memory");
      tdm_load_tile(lds_addr + (unsigned)(k & 1) * (BUF_FLOATS * 4u),
                    gbase + (k + 2) * SUBT);
    }
  }
}

// Pass 1: per (32-row block, chunk) wave, compute b = z-driven state over the
// chunk with zero initial state.  b -> ws1[row][chunk] (float2).
__global__ __launch_bounds__(128) void k_partials(const float* __restrict__ Z,
                                                  float* __restrict__ ws1) {
  __shared__ float tile[WPB * 2 * BUF_FLOATS];
  int lane, r0, c0, chunk; unsigned lds_addr; float* ldsw;
  wave_geometry(lane, r0, c0, chunk, lds_addr, tile, ldsw);

  float p = 0.f, v = 0.f;
  run_chunk<false>(Z + (size_t)r0 * COLS + c0, ldsw, lds_addr, lane, p, v, nullptr);

  ((float2*)ws1)[(size_t)(r0 + lane) * CHUNKS + chunk] = make_float2(p, v);
}

// Pass 2: per row, sequentially combine the 64 chunk summaries:
// s_start[c+1] = M^128 * s_start[c] + b[c], with s_start[0] = (Z[r,0], 0).
__global__ __launch_bounds__(256) void k_combine(const float* __restrict__ Z,
                                                 const float* __restrict__ ws1,
                                                 float* __restrict__ ws2) {
  int r = (int)(blockIdx.x * blockDim.x + threadIdx.x);
  if (r >= ROWS) return;
  // Build M^CHUNK_LEN in double (CHUNK_LEN = 128 -> 7 squarings).
  double e00 = 1.0 - 0.85 - 0.005, e01 = 1.0, e10 = -0.005, e11 = 1.0;
#pragma unroll
  for (int i = 0; i < 7; ++i) {
    double t00 = e00 * e00 + e01 * e10;
    double t01 = e00 * e01 + e01 * e11;
    double t10 = e10 * e00 + e11 * e10;
    double t11 = e10 * e01 + e11 * e11;
    e00 = t00; e01 = t01; e10 = t10; e11 = t11;
  }
  double p = (double)Z[(size_t)r * COLS];
  double v = 0.0;
  const float2* b = (const float2*)ws1 + (size_t)r * CHUNKS;
  float2* s = (float2*)ws2 + (size_t)r * CHUNKS;
  for (int c = 0; c < CHUNKS; ++c) {
    s[c] = make_float2((float)p, (float)v);
    float2 bc = b[c];
    double pn = e00 * p + e01 * v + (double)bc.x;
    double vn = e10 * p + e11 * v + (double)bc.y;
    p = pn; v = vn;
  }
}

// Pass 3: rerun each chunk from its correct start state, emitting x.
__global__ __launch_bounds__(128) void k_final(const float* __restrict__ Z,
                                               const float* __restrict__ ws2,
                                               float* __restrict__ out) {
  __shared__ float tile[WPB * 2 * BUF_FLOATS];
  int lane, r0, c0, chunk; unsigned lds_addr; float* ldsw;
  wave_geometry(lane, r0, c0, chunk, lds_addr, tile, ldsw);

  float2 s = ((const float2*)ws2)[(size_t)(r0 + lane) * CHUNKS + chunk];
  float p = s.x, v = s.y;
  float* optr = out + (size_t)(r0 + lane) * COLS + c0;

  run_chunk<true>(Z + (size_t)r0 * COLS + c0, ldsw, lds_addr, lane, p, v, optr);
}

// Deterministic fallback if workspace is too small: one lane per row.
__global__ void k_naive(const float* __restrict__ Z, float* __restrict__ out) {
  int r = (int)(blockIdx.x * blockDim.x + threadIdx.x);
  if (r >= ROWS) return;
  const float* z = Z + (size_t)r * COLS;
  float* o = out + (size_t)r * COLS;
  float p = z[0], v = 0.f;
  for (int t = 0; t < COLS; ++t) {
    float diff = z[t] - p;
    float x = fmaf(kA, diff, p);
    v = fmaf(kB, diff, v);
    o[t] = x;
    p = x + v;
  }
}

extern "C" void kernel_launch(void* const* d_in, const int* in_sizes, int n_in,
                              void* d_out, int out_size, void* d_ws, size_t ws_size,
                              hipStream_t stream) {
  (void)in_sizes; (void)n_in; (void)out_size;
  const float* Z = (const float*)d_in[0];
  float* out = (float*)d_out;

  const size_t summ_floats = (size_t)ROWS * CHUNKS * 2;   // 0.5M floats = 2MB
  const size_t need = 2 * summ_floats * sizeof(float);    // 4MB scratch

  if (ws_size < need) {
    k_naive<<<(ROWS + 255) / 256, 256, 0, stream>>>(Z, out);
    return;
  }
  float* ws1 = (float*)d_ws;
  float* ws2 = ws1 + summ_floats;

  const int nblk = (ROWBLOCKS * CHUNKS) / WPB;  // 2048 blocks of 4 waves
  k_partials<<<nblk, WPB * 32, 0, stream>>>(Z, ws1);
  k_combine<<<(ROWS + 255) / 256, 256, 0, stream>>>(Z, ws1, ws2);
  k_final<<<nblk, WPB * 32, 0, stream>>>(Z, ws2, out);
}